// GCN_26190710571262
// MI455X (gfx1250) — compile-verified
//
#include <hip/hip_runtime.h>

typedef float v2f __attribute__((ext_vector_type(2)));
typedef float v8f __attribute__((ext_vector_type(8)));

#define D      50
#define KCAT   150   // 3*D concat width
#define KPAD   152   // K padded to multiple of 4 (38 WMMA steps)
#define KLDS   153   // LDS row stride (odd -> conflict-free across 64 banks)
#define WT_ROWS 160
#define WT_COLS 64
#define WAVES_PER_BLK 2

// ---- native fp32 atomics at device scope (resolve in L2 atomic units) ----
__device__ __forceinline__ void atom_add_f32(float* p, float v) {
  asm volatile("global_atomic_add_f32 %0, %1, off scope:SCOPE_DEV"
               :: "v"(p), "v"(v) : "memory");
}
__device__ __forceinline__ void atom_max_f32(float* p, float v) {
  asm volatile("global_atomic_max_num_f32 %0, %1, off scope:SCOPE_DEV"
               :: "v"(p), "v"(v) : "memory");
}

// ---- init workspace: cnt=0, sum=0, max=-FLT_MAX ----
__global__ void gcn_init(float* cnt, float* sum, float* mx, int n_nodes) {
  int i = blockIdx.x * blockDim.x + threadIdx.x;
  int tot = n_nodes * D;
  if (i < tot) { sum[i] = 0.f; mx[i] = -3.402823466e38f; }
  if (i < n_nodes) cnt[i] = 0.f;
}

// ---- build zero-padded W^T in ws: Wt[k][n] = W[n][k], k<160, n<64 ----
__global__ void gcn_build_wt(const float* __restrict__ W, float* __restrict__ Wt) {
  int i = blockIdx.x * blockDim.x + threadIdx.x;
  if (i >= WT_ROWS * WT_COLS) return;
  int k = i >> 6, n = i & 63;
  Wt[i] = (k < KCAT && n < D) ? W[n * KCAT + k] : 0.f;
}

// ---- edge scatter: one wave per edge, lane covers features {lane, lane+32} ----
__global__ void gcn_edge(const float* __restrict__ feat, const float* __restrict__ ew,
                         const int* __restrict__ src, const int* __restrict__ dst,
                         float* cnt, float* sum, float* mx, int n_edges) {
  int lane = threadIdx.x & 31;
  int e = blockIdx.x * (blockDim.x >> 5) + (threadIdx.x >> 5);
  if (e >= n_edges) return;
  int s = src[e], d = dst[e];
  float w = ew[e];
  const float* fs = feat + (size_t)s * D;
  float* ss = sum + (size_t)d * D;
  float* ms = mx  + (size_t)d * D;

  float v0 = fs[lane] * w;
  atom_add_f32(ss + lane, v0);
  atom_max_f32(ms + lane, v0);
  int f1 = lane + 32;
  if (f1 < D) {
    float v1 = fs[f1] * w;
    atom_add_f32(ss + f1, v1);
    atom_max_f32(ms + f1, v1);
  }
  if (lane == 0) atom_add_f32(cnt + d, 1.0f);
}

// ---- node GEMM: one wave per 16-node tile; all operands from LDS ----
__global__ __launch_bounds__(64) void gcn_gemm(
    const float* __restrict__ feat, const float* __restrict__ cnt,
    const float* __restrict__ sum, const float* __restrict__ mx,
    const float* __restrict__ Wt, const float* __restrict__ bias,
    float* __restrict__ out, int n_nodes) {
  __shared__ float Bt[WT_ROWS * WT_COLS];            // 40 KB, shared by both waves
  __shared__ float X[WAVES_PER_BLK][16][KLDS];       // 19.1 KB

  int wave = threadIdx.x >> 5;
  int lane = threadIdx.x & 31;
  int tile = blockIdx.x * WAVES_PER_BLK + wave;
  int row0 = tile * 16;

  // stage padded W^T into LDS (coalesced copy, whole block)
  for (int i = threadIdx.x; i < WT_ROWS * WT_COLS; i += 64)
    Bt[i] = Wt[i];

  // stage concat [feature | mean | max | 0-pad] for this wave's 16 nodes
  for (int r = 0; r < 16; ++r) {
    int node = row0 + r;
    for (int c = lane; c < KLDS; c += 32) {
      float v = 0.f;
      if (node < n_nodes) {
        if (c < D) {
          v = feat[(size_t)node * D + c];
        } else if (c < 2 * D) {
          float cn = cnt[node];
          v = (cn > 0.f) ? sum[(size_t)node * D + (c - D)] / cn : 0.f;
        } else if (c < 3 * D) {
          float cn = cnt[node];
          v = (cn > 0.f) ? mx[(size_t)node * D + (c - 2 * D)] : 0.f;
        }
      }
      X[wave][r][c] = v;
    }
  }
  __syncthreads();

  int half = lane >> 4;   // A: low lanes hold K={0,1}, high lanes K={2,3}
  int lr = lane & 15;
  v8f acc0 = {}, acc1 = {}, acc2 = {}, acc3 = {};

  for (int k0 = 0; k0 < KPAD; k0 += 4) {
    int ka = k0 + half * 2;
    v2f a;
    a.x = X[wave][lr][ka];
    a.y = X[wave][lr][ka + 1];
    // B from LDS: row pair (ka, ka+1); pairs are 64 elements apart -> ds_load_2addr
    const float* b0p = &Bt[ka * WT_COLS + lr];
    v2f b0, b1, b2, b3;
    b0.x = b0p[0];        b0.y = b0p[WT_COLS];
    b1.x = b0p[16];       b1.y = b0p[WT_COLS + 16];
    b2.x = b0p[32];       b2.y = b0p[WT_COLS + 32];
    b3.x = b0p[48];       b3.y = b0p[WT_COLS + 48];
    acc0 = __builtin_amdgcn_wmma_f32_16x16x4_f32(false, a, false, b0, (short)0, acc0, false, false);
    acc1 = __builtin_amdgcn_wmma_f32_16x16x4_f32(false, a, false, b1, (short)0, acc1, false, false);
    acc2 = __builtin_amdgcn_wmma_f32_16x16x4_f32(false, a, false, b2, (short)0, acc2, false, false);
    acc3 = __builtin_amdgcn_wmma_f32_16x16x4_f32(false, a, false, b3, (short)0, acc3, false, false);
  }

  // C layout: VGPR v -> M = v (lanes 0-15) / v+8 (lanes 16-31), N = lane&15
  v8f accs[4] = {acc0, acc1, acc2, acc3};
  int mbase = row0 + half * 8;

  if (row0 + 16 <= n_nodes) {
    // fast path: all 16 rows valid (always taken when n_nodes % 16 == 0)
#pragma unroll
    for (int nt = 0; nt < 4; ++nt) {
      int c = nt * 16 + lr;
      if (c < D) {
        float bb = bias[c];
#pragma unroll
        for (int v = 0; v < 8; ++v) {
          float o = accs[nt][v] + bb;
          out[(size_t)(mbase + v) * D + c] = o > 0.f ? o : 0.f;
        }
      }
    }
  } else {
#pragma unroll
    for (int nt = 0; nt < 4; ++nt) {
      int c = nt * 16 + lr;
      if (c >= D) continue;
      float bb = bias[c];
#pragma unroll
      for (int v = 0; v < 8; ++v) {
        int rowm = mbase + v;
        if (rowm < n_nodes) {
          float o = accs[nt][v] + bb;
          out[(size_t)rowm * D + c] = o > 0.f ? o : 0.f;
        }
      }
    }
  }
}

extern "C" void kernel_launch(void* const* d_in, const int* in_sizes, int n_in,
                              void* d_out, int out_size, void* d_ws, size_t ws_size,
                              hipStream_t stream) {
  const float* feature = (const float*)d_in[0];
  const float* edge_w  = (const float*)d_in[1];
  const int*   src     = (const int*)d_in[2];
  const int*   dst     = (const int*)d_in[3];
  const float* W       = (const float*)d_in[4];
  const float* b       = (const float*)d_in[5];
  float* out = (float*)d_out;

  int n_nodes = in_sizes[0] / D;
  int n_edges = in_sizes[1];

  // workspace layout (floats): cnt[N] | sum[N*D] | mx[N*D] | Wt[160*64]  (~40.5 MB)
  float* ws  = (float*)d_ws;
  float* cnt = ws;
  float* sum = cnt + n_nodes;
  float* mx  = sum + (size_t)n_nodes * D;
  float* Wt  = mx  + (size_t)n_nodes * D;

  int initN = n_nodes * D;
  gcn_init<<<(initN + 255) / 256, 256, 0, stream>>>(cnt, sum, mx, n_nodes);
  gcn_build_wt<<<(WT_ROWS * WT_COLS + 255) / 256, 256, 0, stream>>>(W, Wt);

  int edges_per_block = 256 / 32;
  gcn_edge<<<(n_edges + edges_per_block - 1) / edges_per_block, 256, 0, stream>>>(
      feature, edge_w, src, dst, cnt, sum, mx, n_edges);

  int tiles = (n_nodes + 15) / 16;
  gcn_gemm<<<(tiles + WAVES_PER_BLK - 1) / WAVES_PER_BLK, 64, 0, stream>>>(
      feature, cnt, sum, mx, Wt, b, out, n_nodes);
}